// LundWeight_74491912782168
// MI455X (gfx1250) — compile-verified
//
#include <hip/hip_runtime.h>
#include <stdint.h>

// ---------------------------------------------------------------------------
// LundWeight for MI455X (gfx1250, wave32)
//   grid  = B blocks (one per batch element)
//   block = 256 threads (8 wave32)
// Per block:
//   1. async-copy z[b,:,:] (128*17 floats = 8704 B) global->LDS, b128 chunks
//      (2 unconditional issues + 1 predicated; no dynamic loop)
//   2. threads 0..127 build a per-m coefficient table (all z_max math hoisted)
//   3. all threads evaluate the likelihood ratio per element in log2 space,
//      fully unrolled, (m,k) tracked incrementally (no per-element division)
//   4. product/sum reduction: wave shuffle butterfly + LDS cross-wave
// ---------------------------------------------------------------------------

#define LUND_LOG2E 1.4426950408889634f
#define LUND_CLIP2 (10.0f * LUND_LOG2E)   // EXPMAX in log2 units
#define LUND_OVER  15.0f                   // OVER_SAMPLE
#define LUND_A_BASE 0.72f
#define LUND_B_BASE 0.88f

static constexpr int LUND_M  = 128;
static constexpr int LUND_K  = 17;
static constexpr int LUND_MK = LUND_M * LUND_K;   // 2176
static constexpr int LUND_NT = 256;               // 8 waves

// --- native-rate transcendental helpers (v_log_f32 is log2, v_exp_f32 is exp2)
__device__ __forceinline__ float fast_log2(float x) {
#if defined(__HIP_DEVICE_COMPILE__) && __has_builtin(__builtin_amdgcn_logf)
  return __builtin_amdgcn_logf(x);
#else
  return log2f(x);
#endif
}
__device__ __forceinline__ float fast_exp2(float x) {
#if defined(__HIP_DEVICE_COMPILE__) && __has_builtin(__builtin_amdgcn_exp2f)
  return __builtin_amdgcn_exp2f(x);
#else
  return exp2f(x);
#endif
}
__device__ __forceinline__ float fast_rcp(float x) {
#if defined(__HIP_DEVICE_COMPILE__) && __has_builtin(__builtin_amdgcn_rcpf)
  return __builtin_amdgcn_rcpf(x);
#else
  return 1.0f / x;
#endif
}

// All z-independent pieces of log f(z) for one (a, b, mT), in log2 units:
//   F(z) = C0 - BX*(1/z) - log2(z) + aeff*log2(1-z)
// where C0 = BX/z_max + log2(z_max) - aeff*log2(1-z_max), BX = b*mT^2*log2(e)
__device__ __forceinline__ void lund_coef(float a, float bprm, float mt,
                                          float& C0, float& BX) {
  const float c = 1.0f;
  float bexp = bprm * mt * mt;
  bool a0 = (a < 0.02f);                 // AFROMZERO
  bool ac = (fabsf(a - c) < 0.01f);      // AFROMC
  float denom = (a0 || ac) ? 1.0f : (c - a);
  float t = bexp - c;
  float disc = sqrtf(t * t + 4.0f * a * bexp);
  float zgen = 0.5f * (bexp + c - disc) / denom;
  if ((zgen > 0.9999f) && (bexp > 100.0f))
    zgen = fminf(zgen, 1.0f - a / bexp);
  float zmax = a0 ? ((c > bexp) ? (bexp / c) : 1.0f)
                  : (ac ? (bexp / (bexp + c)) : zgen);
  float aeff   = a0 ? 0.0f : a;
  float izm    = 1.0f / zmax;
  float l2zm   = fast_log2(zmax);
  float l2omzm = a0 ? 0.0f : fast_log2(1.0f - zmax);  // guarded: zmax may be 1
  BX = bexp * LUND_LOG2E;
  C0 = BX * izm + l2zm - aeff * l2omzm;
}

__device__ __forceinline__ void async_b128(unsigned lds_addr,
                                           unsigned long long gaddr) {
#if defined(__HIP_DEVICE_COMPILE__)
  asm volatile("global_load_async_to_lds_b128 %0, %1, off"
               :: "v"(lds_addr), "v"(gaddr) : "memory");
#endif
}

__global__ __launch_bounds__(LUND_NT)
void LundWeight_kernel(const float* __restrict__ z,
                       const float* __restrict__ mT,
                       const int*   __restrict__ obsArr,
                       const float* __restrict__ pa,
                       const float* __restrict__ pb,
                       float*       __restrict__ out) {
  __shared__ __align__(16) float  zsh[LUND_MK];      // 8704 B tile of z[b]
  __shared__ __align__(16) float4 coef[LUND_M];      // per-m coefficients
  __shared__ float sS[LUND_NT / 32];
  __shared__ float sP[LUND_NT / 32];

  const int b = blockIdx.x;
  const int t = threadIdx.x;

  // ---- 1. async global->LDS copy of this block's z tile -------------------
  // 544 x b128 chunks: 2 full rounds of 256 lanes + 32-lane remainder.
  {
    const float* gbase = z + (size_t)b * LUND_MK;
    unsigned lbase = (unsigned)(uintptr_t)(&zsh[0]) + (unsigned)(t * 16);
    unsigned long long gb64 = (unsigned long long)(uintptr_t)(gbase) +
                              (unsigned long long)(t * 16);
    async_b128(lbase,          gb64);
    async_b128(lbase + 4096u,  gb64 + 4096ull);
    if (t < (LUND_MK / 4 - 2 * LUND_NT))      // t < 32
      async_b128(lbase + 8192u, gb64 + 8192ull);
  }

  // ---- 2. per-m coefficient table (overlaps the async copy) ---------------
  const float a_new = pa[0];
  const float b_new = pb[0];
  if (t < LUND_M) {
    float mt = mT[(size_t)b * LUND_M + t];
    float C0n, BXn, C0b, BXb;
    lund_coef(a_new, b_new, mt, C0n, BXn);
    lund_coef(LUND_A_BASE, LUND_B_BASE, mt, C0b, BXb);
    coef[t] = make_float4(C0n, BXn, C0b, BXb);
  }

  const float an_eff = (a_new < 0.02f) ? 0.0f : a_new;
  const float ab_eff = LUND_A_BASE;                 // 0.72 never hits a==0/a==c
  const int   obs    = obsArr[b];                   // block-uniform -> SGPR

  // wait for our wave's async copies, then cross-wave barrier
#if defined(__HIP_DEVICE_COMPILE__) && __has_builtin(__builtin_amdgcn_s_wait_asynccnt)
  __builtin_amdgcn_s_wait_asynccnt(0);
#elif defined(__HIP_DEVICE_COMPILE__)
  asm volatile("s_wait_asynccnt 0" ::: "memory");
#endif
  __syncthreads();

  // ---- 3. per-element likelihood ratio, log2 space ------------------------
  float S = 0.0f;   // sum of (Fn - Fb) over accepted (k==0) elements
  float P = 1.0f;   // product of (15-gn)/(15-gb) over rejected (k>0) elements

  int m = t / LUND_K;            // one division at entry; then incremental
  int k = t - m * LUND_K;

  auto body = [&](int e, int mm, int kk) {
    float zv   = zsh[e];
    bool  mask = (mm < obs) && (zv != 0.0f);
    float zz   = mask ? zv : 0.5f;                  // z_safe as in reference

    float rz = fast_rcp(zz);
    float lz = fast_log2(zz);
    float l1 = fast_log2(1.0f - zz);
    float4 cf = coef[mm];

    float Fn = cf.x - cf.y * rz - lz + an_eff * l1;
    float Fb = cf.z - cf.w * rz - lz + ab_eff * l1;
    Fn = fminf(fmaxf(Fn, -LUND_CLIP2), LUND_CLIP2);
    Fb = fminf(fmaxf(Fb, -LUND_CLIP2), LUND_CLIP2);

    float gn = fast_exp2(Fn);
    float gb = fast_exp2(Fb);
    float w  = (LUND_OVER - gn) * fast_rcp(LUND_OVER - gb);

    S += (mask && (kk == 0)) ? (Fn - Fb) : 0.0f;
    P *= (mask && (kk != 0)) ? w : 1.0f;
  };

  // stride 256 = 15*17 + 1  =>  m += 15, k += 1 (with carry) per step
#pragma unroll
  for (int i = 0; i < 8; ++i) {                // e = t .. t+1792, always valid
    body(t + i * LUND_NT, m, k);
    k += 1; m += 15;
    if (k >= LUND_K) { k -= LUND_K; m += 1; }
  }
  if (t < (LUND_MK - 8 * LUND_NT))             // tail: e = t+2048, t < 128
    body(t + 8 * LUND_NT, m, k);

  // ---- 4. reduction: wave32 butterfly, then LDS across 8 waves ------------
#pragma unroll
  for (int off = 16; off > 0; off >>= 1) {
    S += __shfl_xor(S, off, 32);
    P *= __shfl_xor(P, off, 32);
  }
  int wid = t >> 5;
  if ((t & 31) == 0) { sS[wid] = S; sP[wid] = P; }
  __syncthreads();
  if (t == 0) {
    float St = 0.0f, Pt = 1.0f;
#pragma unroll
    for (int i = 0; i < LUND_NT / 32; ++i) { St += sS[i]; Pt *= sP[i]; }
    out[b] = fast_exp2(St) * Pt;
  }
}

extern "C" void kernel_launch(void* const* d_in, const int* in_sizes, int n_in,
                              void* d_out, int out_size, void* d_ws, size_t ws_size,
                              hipStream_t stream) {
  const float* z   = (const float*)d_in[0];   // (B, 128, 17) f32
  const float* mT  = (const float*)d_in[1];   // (B, 128)     f32
  const int*   obs = (const int*)  d_in[2];   // (B,)         i32
  const float* pa  = (const float*)d_in[3];   // scalar f32
  const float* pb  = (const float*)d_in[4];   // scalar f32
  float* out = (float*)d_out;                 // (B,) f32
  const int B = out_size;                     // 8192
  (void)in_sizes; (void)n_in; (void)d_ws; (void)ws_size;
  LundWeight_kernel<<<B, LUND_NT, 0, stream>>>(z, mT, obs, pa, pb, out);
}